// DeeperGCN_40733469835822
// MI455X (gfx1250) — compile-verified
//
#include <hip/hip_runtime.h>
#include <hip/hip_bf16.h>

#define N_NODES 10000
#define N_EDGES 160000
#define NUM_FEAT 14
#define EDGE_DIM 4
#define DIM 64
#define HID 128
#define LAYERS 28
#define MSG_EPS 1e-7f
#define LN_EPS 1e-5f

typedef float v2f __attribute__((ext_vector_type(2)));
typedef float v8f __attribute__((ext_vector_type(8)));

// ---------- helpers ----------

__device__ __forceinline__ float wave_sum(float v) {
    #pragma unroll
    for (int off = 16; off > 0; off >>= 1)
        v += __shfl_xor(v, off, 32);
    return v;
}

// order-preserving float<->uint mapping for atomic max on positive/negative floats
__device__ __forceinline__ unsigned f2ord(float f) {
    unsigned u = __float_as_uint(f);
    return (u & 0x80000000u) ? ~u : (u | 0x80000000u);
}
__device__ __forceinline__ float ord2f(unsigned u) {
    return (u & 0x80000000u) ? __uint_as_float(u & 0x7fffffffu) : __uint_as_float(~u);
}
#define ORD_NEG_INF 0x007FFFFFu

__device__ __forceinline__ v8f wmma4(v2f a, v2f b, v8f c) {
    // V_WMMA_F32_16X16X4_F32 : D = A(16x4) * B(4x16) + C(16x16), all f32
    return __builtin_amdgcn_wmma_f32_16x16x4_f32(false, a, false, b, (short)0, c, false, false);
}

// ---------- embeddings ----------

__global__ __launch_bounds__(256) void node_embed_kernel(
    const float* __restrict__ x, const float* __restrict__ Wn,
    const float* __restrict__ bn, float* __restrict__ u)
{
    int tid = blockIdx.x * blockDim.x + threadIdx.x;
    if (tid >= N_NODES * DIM) return;
    int n = tid / DIM, c = tid % DIM;
    float acc = bn[c];
    #pragma unroll
    for (int k = 0; k < NUM_FEAT; ++k)
        acc = fmaf(x[n * NUM_FEAT + k], Wn[k * DIM + c], acc);
    u[tid] = acc;
}

__global__ __launch_bounds__(256) void edge_embed_kernel(
    const float* __restrict__ edge_attr, const float* __restrict__ We,
    const float* __restrict__ be, float* __restrict__ ea)
{
    int tid = blockIdx.x * blockDim.x + threadIdx.x;
    if (tid >= N_EDGES * DIM) return;
    int e = tid / DIM, c = tid % DIM;
    float acc = be[c];
    #pragma unroll
    for (int k = 0; k < EDGE_DIM; ++k)
        acc = fmaf(edge_attr[e * EDGE_DIM + k], We[k * DIM + c], acc);
    ea[tid] = acc;
}

// ---------- per-layer: pre layernorm+relu (one wave per node) ----------

__global__ __launch_bounds__(128) void pre_ln_kernel(
    const float* __restrict__ hc, const float* __restrict__ ln_g,
    const float* __restrict__ ln_b, int layer, float* __restrict__ u)
{
    int node = blockIdx.x * (blockDim.x >> 5) + (threadIdx.x >> 5);
    int lane = threadIdx.x & 31;
    if (node >= N_NODES) return;
    const float* g = ln_g + layer * DIM;
    const float* b = ln_b + layer * DIM;
    float v0 = hc[node * DIM + lane];
    float v1 = hc[node * DIM + lane + 32];
    float mu = wave_sum(v0 + v1) * (1.f / 64.f);
    float d0 = v0 - mu, d1 = v1 - mu;
    float var = wave_sum(d0 * d0 + d1 * d1) * (1.f / 64.f);
    float rs = rsqrtf(var + LN_EPS);
    float o0 = fmaf(d0 * rs, g[lane], b[lane]);
    float o1 = fmaf(d1 * rs, g[lane + 32], b[lane + 32]);
    u[node * DIM + lane]      = fmaxf(o0, 0.f);
    u[node * DIM + lane + 32] = fmaxf(o1, 0.f);
}

// ---------- init reduction buffers ----------

__global__ __launch_bounds__(256) void init_kernel(
    unsigned* __restrict__ m_ord, float* __restrict__ s, float* __restrict__ agg)
{
    int tid = blockIdx.x * blockDim.x + threadIdx.x;
    if (tid >= N_NODES * DIM) return;
    m_ord[tid] = ORD_NEG_INF;
    s[tid] = 0.f;
    agg[tid] = 0.f;
}

// ---------- edge pass 1: segment max of logits ----------

__global__ __launch_bounds__(256) void edge_max_kernel(
    const float* __restrict__ u, const float* __restrict__ ea,
    const int* __restrict__ src, const int* __restrict__ dst,
    const float* __restrict__ t_arr, int layer, unsigned* __restrict__ m_ord)
{
    int tid = blockIdx.x * blockDim.x + threadIdx.x;
    if (tid >= N_EDGES * (DIM / 4)) return;
    int e = tid >> 4;
    int c = (tid & 15) * 4;
    float t = t_arr[layer];
    int sn = src[e], dn = dst[e];
    const float4 uv = *reinterpret_cast<const float4*>(u + sn * DIM + c);
    const float4 ev = *reinterpret_cast<const float4*>(ea + e * DIM + c);
    unsigned* mp = m_ord + dn * DIM + c;
    float l0 = (fmaxf(uv.x + ev.x, 0.f) + MSG_EPS) * t;
    float l1 = (fmaxf(uv.y + ev.y, 0.f) + MSG_EPS) * t;
    float l2 = (fmaxf(uv.z + ev.z, 0.f) + MSG_EPS) * t;
    float l3 = (fmaxf(uv.w + ev.w, 0.f) + MSG_EPS) * t;
    atomicMax(mp + 0, f2ord(l0));
    atomicMax(mp + 1, f2ord(l1));
    atomicMax(mp + 2, f2ord(l2));
    atomicMax(mp + 3, f2ord(l3));
}

// ---------- edge pass 2: e = exp(logit-m); accumulate s and msg*e ----------

__global__ __launch_bounds__(256) void edge_sum_kernel(
    const float* __restrict__ u, const float* __restrict__ ea,
    const int* __restrict__ src, const int* __restrict__ dst,
    const float* __restrict__ t_arr, int layer,
    const unsigned* __restrict__ m_ord, float* __restrict__ s, float* __restrict__ agg)
{
    int tid = blockIdx.x * blockDim.x + threadIdx.x;
    if (tid >= N_EDGES * (DIM / 4)) return;
    int e = tid >> 4;
    int c = (tid & 15) * 4;
    float t = t_arr[layer];
    int sn = src[e], dn = dst[e];
    const float4 uv = *reinterpret_cast<const float4*>(u + sn * DIM + c);
    const float4 ev = *reinterpret_cast<const float4*>(ea + e * DIM + c);
    const unsigned* mp = m_ord + dn * DIM + c;
    float* sp = s + dn * DIM + c;
    float* ap = agg + dn * DIM + c;
    float msg[4], lg[4];
    msg[0] = fmaxf(uv.x + ev.x, 0.f) + MSG_EPS;
    msg[1] = fmaxf(uv.y + ev.y, 0.f) + MSG_EPS;
    msg[2] = fmaxf(uv.z + ev.z, 0.f) + MSG_EPS;
    msg[3] = fmaxf(uv.w + ev.w, 0.f) + MSG_EPS;
    #pragma unroll
    for (int j = 0; j < 4; ++j) {
        lg[j] = msg[j] * t;
        float ex = expf(lg[j] - ord2f(mp[j]));
        atomicAdd(sp + j, ex);
        atomicAdd(ap + j, msg[j] * ex);
    }
}

// ---------- per-layer node MLP with WMMA (one wave per 16 nodes) ----------

__global__ __launch_bounds__(32) void mlp_wmma_kernel(
    const float* __restrict__ u, const float* __restrict__ agg, const float* __restrict__ s,
    const float* __restrict__ W1_all, const float* __restrict__ b1_all,
    const float* __restrict__ g1_all, const float* __restrict__ be1_all,
    const float* __restrict__ W2_all, const float* __restrict__ b2_all,
    int layer, int add_residual, float* __restrict__ hc)
{
    __shared__ float hmid[16][DIM];       // GEMM1 A operand
    __shared__ float z[16][HID];          // GEMM1 out / GEMM2 A operand

    const float* W1  = W1_all  + (size_t)layer * DIM * HID;
    const float* b1  = b1_all  + layer * HID;
    const float* g1  = g1_all  + layer * HID;
    const float* be1 = be1_all + layer * HID;
    const float* W2  = W2_all  + (size_t)layer * HID * DIM;
    const float* b2  = b2_all  + layer * DIM;

    int tile = blockIdx.x;                // 625 tiles of 16 nodes
    int lane = threadIdx.x;
    int lm   = lane & 15;                 // M index (A) or N index (B/C)
    int half = lane >> 4;                 // 0: K pair {0,1}, 1: K pair {2,3}
    int koff = half * 2;

    // hmid = agg/(s+1e-16) + u  (the gen_conv residual: h = agg + x)
    for (int idx = lane; idx < 16 * DIM; idx += 32) {
        int r = idx >> 6, c = idx & 63;
        int node = tile * 16 + r;
        float a  = agg[node * DIM + c];
        float ss = s[node * DIM + c];
        hmid[r][c] = a / (ss + 1e-16f) + u[node * DIM + c];
    }
    __syncthreads();

    // GEMM1: z = hmid(16x64) @ W1(64x128) + b1    (8 N-tiles x 16 WMMA each)
    for (int nt = 0; nt < 8; ++nt) {
        float bias = b1[nt * 16 + lm];
        v8f acc = {bias, bias, bias, bias, bias, bias, bias, bias};
        #pragma unroll
        for (int k0 = 0; k0 < DIM; k0 += 4) {
            v2f a = {hmid[lm][k0 + koff], hmid[lm][k0 + koff + 1]};
            v2f b = {W1[(k0 + koff) * HID + nt * 16 + lm],
                     W1[(k0 + koff + 1) * HID + nt * 16 + lm]};
            acc = wmma4(a, b, acc);
        }
        #pragma unroll
        for (int r = 0; r < 8; ++r)
            z[r + half * 8][nt * 16 + lm] = acc[r];
    }
    __syncthreads();

    // layernorm over HID=128 per row + relu, in place in z
    for (int r = 0; r < 16; ++r) {
        float x0 = z[r][lane], x1 = z[r][lane + 32];
        float x2 = z[r][lane + 64], x3 = z[r][lane + 96];
        float mu = wave_sum(x0 + x1 + x2 + x3) * (1.f / 128.f);
        float d0 = x0 - mu, d1 = x1 - mu, d2 = x2 - mu, d3 = x3 - mu;
        float var = wave_sum(d0 * d0 + d1 * d1 + d2 * d2 + d3 * d3) * (1.f / 128.f);
        float rs = rsqrtf(var + LN_EPS);
        z[r][lane]      = fmaxf(fmaf(d0 * rs, g1[lane],      be1[lane]),      0.f);
        z[r][lane + 32] = fmaxf(fmaf(d1 * rs, g1[lane + 32], be1[lane + 32]), 0.f);
        z[r][lane + 64] = fmaxf(fmaf(d2 * rs, g1[lane + 64], be1[lane + 64]), 0.f);
        z[r][lane + 96] = fmaxf(fmaf(d3 * rs, g1[lane + 96], be1[lane + 96]), 0.f);
    }
    __syncthreads();

    // GEMM2: out = z(16x128) @ W2(128x64) + b2; hc = (residual?) + out
    for (int nt = 0; nt < 4; ++nt) {
        float bias = b2[nt * 16 + lm];
        v8f acc = {bias, bias, bias, bias, bias, bias, bias, bias};
        #pragma unroll
        for (int k0 = 0; k0 < HID; k0 += 4) {
            v2f a = {z[lm][k0 + koff], z[lm][k0 + koff + 1]};
            v2f b = {W2[(k0 + koff) * DIM + nt * 16 + lm],
                     W2[(k0 + koff + 1) * DIM + nt * 16 + lm]};
            acc = wmma4(a, b, acc);
        }
        #pragma unroll
        for (int r = 0; r < 8; ++r) {
            int node = tile * 16 + r + half * 8;
            int col  = nt * 16 + lm;
            float base = add_residual ? hc[node * DIM + col] : 0.f;
            hc[node * DIM + col] = base + acc[r];
        }
    }
}

// ---------- final: relu(LN(hc, ln_g[0], ln_b[0])) @ W_out + b_out ----------

__global__ __launch_bounds__(32) void final_wmma_kernel(
    const float* __restrict__ hc, const float* __restrict__ ln_g,
    const float* __restrict__ ln_b, const float* __restrict__ W_out,
    const float* __restrict__ b_out, float* __restrict__ out)
{
    __shared__ float y[16][DIM];
    int tile = blockIdx.x;
    int lane = threadIdx.x;
    int lm   = lane & 15;
    int half = lane >> 4;
    int koff = half * 2;

    for (int r = 0; r < 16; ++r) {
        int node = tile * 16 + r;
        float v0 = hc[node * DIM + lane];
        float v1 = hc[node * DIM + lane + 32];
        float mu = wave_sum(v0 + v1) * (1.f / 64.f);
        float d0 = v0 - mu, d1 = v1 - mu;
        float var = wave_sum(d0 * d0 + d1 * d1) * (1.f / 64.f);
        float rs = rsqrtf(var + LN_EPS);
        y[r][lane]      = fmaxf(fmaf(d0 * rs, ln_g[lane],      ln_b[lane]),      0.f);
        y[r][lane + 32] = fmaxf(fmaf(d1 * rs, ln_g[lane + 32], ln_b[lane + 32]), 0.f);
    }
    __syncthreads();

    for (int nt = 0; nt < 4; ++nt) {
        float bias = b_out[nt * 16 + lm];
        v8f acc = {bias, bias, bias, bias, bias, bias, bias, bias};
        #pragma unroll
        for (int k0 = 0; k0 < DIM; k0 += 4) {
            v2f a = {y[lm][k0 + koff], y[lm][k0 + koff + 1]};
            v2f b = {W_out[(k0 + koff) * DIM + nt * 16 + lm],
                     W_out[(k0 + koff + 1) * DIM + nt * 16 + lm]};
            acc = wmma4(a, b, acc);
        }
        #pragma unroll
        for (int r = 0; r < 8; ++r) {
            int node = tile * 16 + r + half * 8;
            out[node * DIM + nt * 16 + lm] = acc[r];
        }
    }
}

// ---------- host launch ----------

extern "C" void kernel_launch(void* const* d_in, const int* in_sizes, int n_in,
                              void* d_out, int out_size, void* d_ws, size_t ws_size,
                              hipStream_t stream) {
    const float* x         = (const float*)d_in[0];
    const float* edge_attr = (const float*)d_in[1];
    const int*   src       = (const int*)d_in[2];
    const int*   dst       = (const int*)d_in[3];
    const float* W_node    = (const float*)d_in[4];
    const float* b_node    = (const float*)d_in[5];
    const float* W_edge    = (const float*)d_in[6];
    const float* b_edge    = (const float*)d_in[7];
    const float* t_arr     = (const float*)d_in[8];
    const float* W1        = (const float*)d_in[9];
    const float* b1        = (const float*)d_in[10];
    const float* g1        = (const float*)d_in[11];
    const float* be1       = (const float*)d_in[12];
    const float* W2        = (const float*)d_in[13];
    const float* b2        = (const float*)d_in[14];
    const float* ln_g      = (const float*)d_in[15];
    const float* ln_b      = (const float*)d_in[16];
    const float* W_out     = (const float*)d_in[17];
    const float* b_out     = (const float*)d_in[18];
    float* out = (float*)d_out;

    // workspace layout (floats)
    float* ws = (float*)d_ws;
    float*    ea    = ws;                                  // N_EDGES*DIM
    float*    hc    = ea + (size_t)N_EDGES * DIM;          // N_NODES*DIM
    float*    u     = hc + (size_t)N_NODES * DIM;          // N_NODES*DIM
    unsigned* m_ord = (unsigned*)(u + (size_t)N_NODES * DIM); // N_NODES*DIM
    float*    sbuf  = (float*)(m_ord + (size_t)N_NODES * DIM);
    float*    agg   = sbuf + (size_t)N_NODES * DIM;

    const int nodeElems = N_NODES * DIM;                   // 640000
    const int edgeElems = N_EDGES * DIM;                   // 10240000
    const int edgeThreads = N_EDGES * (DIM / 4);           // 2560000
    const int TILES = N_NODES / 16;                        // 625

    // embeddings: u <- node embed (gen_conv input for layer 0), ea <- edge embed
    node_embed_kernel<<<(nodeElems + 255) / 256, 256, 0, stream>>>(x, W_node, b_node, u);
    edge_embed_kernel<<<(edgeElems + 255) / 256, 256, 0, stream>>>(edge_attr, W_edge, b_edge, ea);

    for (int layer = 0; layer < LAYERS; ++layer) {
        if (layer > 0) {
            // u = relu(layernorm(hc, ln_g[layer], ln_b[layer]))
            pre_ln_kernel<<<(N_NODES + 3) / 4, 128, 0, stream>>>(hc, ln_g, ln_b, layer, u);
        }
        init_kernel<<<(nodeElems + 255) / 256, 256, 0, stream>>>(m_ord, sbuf, agg);
        edge_max_kernel<<<(edgeThreads + 255) / 256, 256, 0, stream>>>(
            u, ea, src, dst, t_arr, layer, m_ord);
        edge_sum_kernel<<<(edgeThreads + 255) / 256, 256, 0, stream>>>(
            u, ea, src, dst, t_arr, layer, m_ord, sbuf, agg);
        mlp_wmma_kernel<<<TILES, 32, 0, stream>>>(
            u, agg, sbuf, W1, b1, g1, be1, W2, b2,
            layer, /*add_residual=*/(layer > 0) ? 1 : 0, hc);
    }

    final_wmma_kernel<<<TILES, 32, 0, stream>>>(hc, ln_g, ln_b, W_out, b_out, out);
    (void)in_sizes; (void)n_in; (void)out_size; (void)ws_size;
}